// SyntheticModelNative_23502061043761
// MI455X (gfx1250) — compile-verified
//
#include <hip/hip_runtime.h>

#define NUM_TABLES 26
#define NUM_ROWS   50000
#define WIDTH      128
#define BATCH      4096
#define NNZ        20
#define NUM_NUM    13
#define D_IN       3341          // 26*128 + 13
#define XK         3344          // D_IN padded to multiple of 16

typedef __attribute__((ext_vector_type(2))) float v2f;
typedef __attribute__((ext_vector_type(8))) float v8f;

// ---------------- embedding gather + pool -> X[B, XK] ----------------
__global__ __launch_bounds__(WIDTH) void pool_kernel(
    const float* __restrict__ tables, const int* __restrict__ cat,
    float* __restrict__ X)
{
  const int b = blockIdx.x;      // batch
  const int t = blockIdx.y;      // table
  const int j = threadIdx.x;     // feature column 0..127
  __shared__ int sidx[NNZ];
  if (j < NNZ) sidx[j] = cat[((size_t)t * BATCH + b) * NNZ + j];
  __syncthreads();
  const float* tb = tables + (size_t)t * NUM_ROWS * WIDTH + j;
  float acc = 0.f;
#pragma unroll
  for (int n = 0; n < NNZ; ++n)
    acc += tb[(size_t)sidx[n] * WIDTH];
  X[(size_t)b * XK + t * WIDTH + j] = acc;
}

// ------------- numerical features + zero pad tail of X ---------------
__global__ void tail_kernel(const float* __restrict__ numerical,
                            float* __restrict__ X)
{
  const int b = blockIdx.x;
  const int j = threadIdx.x;                 // 0..15
  const float v = (j < NUM_NUM) ? numerical[(size_t)b * NUM_NUM + j] : 0.f;
  X[(size_t)b * XK + NUM_TABLES * WIDTH + j] = v;
}

// ---------------- zero-pad W1 [3341,512] -> [3344,512] ---------------
__global__ __launch_bounds__(256) void padw1_kernel(const float* __restrict__ W1,
                                                    float* __restrict__ W1p)
{
  const int r = blockIdx.x;                  // 0..3343
  const int c = threadIdx.x * 2;             // 2 cols per thread
  float2 v = make_float2(0.f, 0.f);
  if (r < D_IN) v = *(const float2*)(W1 + (size_t)r * 512 + c);
  *(float2*)(W1p + (size_t)r * 512 + c) = v;
}

// ------------- GEMM: C = act(A[M,K] @ B[K,N] + bias) -----------------
// Block tile 128x128, 8 waves (wave32) in 4(M) x 2(N), each wave 2x4
// WMMA 16x16 tiles using V_WMMA_F32_16X16X4_F32. K staged 16-wide in LDS.
// Both A and B tiles kept fragment-major in LDS so every fragment read is
// one aligned ds_load_b64 into an adjacent VGPR pair (no repack movs).
#define BM 128
#define BN 128
#define BK 16

__global__ __launch_bounds__(256) void gemm_bias_act(
    const float* __restrict__ A, const float* __restrict__ B,
    const float* __restrict__ bias, float* __restrict__ C,
    int N, int K, int relu)
{
  __shared__ float As [BM][BK + 2];   // M-major, +2 pad (b64 align, bank spread)
  __shared__ float Bst[BN][BK + 2];   // N-major (transposed), same padding

  const int tid  = threadIdx.x;
  const int lane = tid & 31;
  const int wave = tid >> 5;
  const int wm   = wave & 3;             // wave M index (0..3) -> rows wm*32..+31
  const int wn   = wave >> 2;            // wave N index (0..1) -> cols wn*64..+63
  const int lm    = lane & 15;
  const int khalf = (lane >> 4) << 1;    // 0 for lanes 0-15, 2 for lanes 16-31

  const int bm = blockIdx.y * BM;
  const int bn = blockIdx.x * BN;

  // global->LDS mapping (float4 per thread, 2 passes each)
  const int a_row = tid >> 2;            // 0..63 (and +64)
  const int a_col = (tid & 3) << 2;      // 0,4,8,12
  const int b_row = tid >> 5;            // 0..7 (and +8)   (K index of W tile)
  const int b_col = (tid & 31) << 2;     // 0..124          (N index of W tile)

  v8f acc[2][4];
#pragma unroll
  for (int i = 0; i < 2; ++i)
#pragma unroll
    for (int j = 0; j < 4; ++j)
#pragma unroll
      for (int r = 0; r < 8; ++r) acc[i][j][r] = 0.f;

  for (int k0 = 0; k0 < K; k0 += BK) {
    const float4 a0 = *(const float4*)(A + (size_t)(bm + a_row)      * K + k0 + a_col);
    const float4 a1 = *(const float4*)(A + (size_t)(bm + a_row + 64) * K + k0 + a_col);
    const float4 w0 = *(const float4*)(B + (size_t)(k0 + b_row)     * N + bn + b_col);
    const float4 w1 = *(const float4*)(B + (size_t)(k0 + b_row + 8) * N + bn + b_col);

    // prefetch next K-stage into cache (lowers to global_prefetch_b8)
    if (k0 + BK < K) {
      __builtin_prefetch(A + (size_t)(bm + a_row)      * K + k0 + BK + a_col, 0, 3);
      __builtin_prefetch(A + (size_t)(bm + a_row + 64) * K + k0 + BK + a_col, 0, 3);
      __builtin_prefetch(B + (size_t)(k0 + BK + b_row)     * N + bn + b_col, 0, 3);
      __builtin_prefetch(B + (size_t)(k0 + BK + b_row + 8) * N + bn + b_col, 0, 3);
    }

    __syncthreads();
    As[a_row][a_col + 0] = a0.x; As[a_row][a_col + 1] = a0.y;
    As[a_row][a_col + 2] = a0.z; As[a_row][a_col + 3] = a0.w;
    As[a_row + 64][a_col + 0] = a1.x; As[a_row + 64][a_col + 1] = a1.y;
    As[a_row + 64][a_col + 2] = a1.z; As[a_row + 64][a_col + 3] = a1.w;
    Bst[b_col + 0][b_row]     = w0.x; Bst[b_col + 1][b_row]     = w0.y;
    Bst[b_col + 2][b_row]     = w0.z; Bst[b_col + 3][b_row]     = w0.w;
    Bst[b_col + 0][b_row + 8] = w1.x; Bst[b_col + 1][b_row + 8] = w1.y;
    Bst[b_col + 2][b_row + 8] = w1.z; Bst[b_col + 3][b_row + 8] = w1.w;
    __syncthreads();

#pragma unroll
    for (int kk = 0; kk < BK; kk += 4) {
      // A fragment 16x4: lanes 0-15 -> K=kk+0,kk+1 ; lanes 16-31 -> K=kk+2,kk+3
      v2f af[2];
#pragma unroll
      for (int i = 0; i < 2; ++i)
        af[i] = *(const v2f*)&As[wm * 32 + i * 16 + lm][kk + khalf];
      // B fragment 4x16: VGPR0 rows {K, K+2}, VGPR1 rows {K+1, K+3} across halves
      v2f bf[4];
#pragma unroll
      for (int j = 0; j < 4; ++j)
        bf[j] = *(const v2f*)&Bst[wn * 64 + j * 16 + lm][kk + khalf];
#pragma unroll
      for (int i = 0; i < 2; ++i)
#pragma unroll
        for (int j = 0; j < 4; ++j)
          acc[i][j] = __builtin_amdgcn_wmma_f32_16x16x4_f32(
              false, af[i], false, bf[j], (short)0, acc[i][j], false, false);
    }
  }

  // epilogue: C/D layout -> lanes 0-15 rows r, lanes 16-31 rows r+8
#pragma unroll
  for (int i = 0; i < 2; ++i) {
    const int row0 = bm + wm * 32 + i * 16 + ((lane >> 4) << 3);
#pragma unroll
    for (int j = 0; j < 4; ++j) {
      const int col = bn + wn * 64 + j * 16 + lm;
      const float bv = bias[col];
#pragma unroll
      for (int r = 0; r < 8; ++r) {
        float v = acc[i][j][r] + bv;
        if (relu) v = fmaxf(v, 0.f);
        C[(size_t)(row0 + r) * N + col] = v;
      }
    }
  }
}

// ---------------- final layer: out[b] = H3[b,:] . W4 + b4 ------------
__global__ __launch_bounds__(256) void final_layer(
    const float* __restrict__ H, const float* __restrict__ W,
    const float* __restrict__ bias, float* __restrict__ out)
{
  const int lane = threadIdx.x & 31;
  const int row  = blockIdx.x * 8 + (threadIdx.x >> 5);
  const float4 h = ((const float4*)(H + (size_t)row * 128))[lane];
  const float4 w = ((const float4*)W)[lane];
  float s = h.x * w.x + h.y * w.y + h.z * w.z + h.w * w.w;
#pragma unroll
  for (int off = 16; off > 0; off >>= 1)
    s += __shfl_xor(s, off, 32);
  if (lane == 0) out[row] = s + bias[0];
}

extern "C" void kernel_launch(void* const* d_in, const int* in_sizes, int n_in,
                              void* d_out, int out_size, void* d_ws, size_t ws_size,
                              hipStream_t stream) {
  const float* numerical = (const float*)d_in[0];
  const int*   cat       = (const int*)  d_in[1];
  const float* tables    = (const float*)d_in[2];
  const float* W1 = (const float*)d_in[3];
  const float* b1 = (const float*)d_in[4];
  const float* W2 = (const float*)d_in[5];
  const float* b2 = (const float*)d_in[6];
  const float* W3 = (const float*)d_in[7];
  const float* b3 = (const float*)d_in[8];
  const float* W4 = (const float*)d_in[9];
  const float* b4 = (const float*)d_in[10];
  float* out = (float*)d_out;

  // workspace carve-up (all sizes multiples of 256 B)
  char* ws = (char*)d_ws;
  float* X   = (float*)ws;  ws += (size_t)BATCH * XK  * 4;   // 54.8 MB
  float* W1p = (float*)ws;  ws += (size_t)XK    * 512 * 4;   //  6.9 MB
  float* H1  = (float*)ws;  ws += (size_t)BATCH * 512 * 4;   //  8.4 MB
  float* H2  = (float*)ws;  ws += (size_t)BATCH * 256 * 4;   //  4.2 MB
  float* H3  = (float*)ws;  ws += (size_t)BATCH * 128 * 4;   //  2.1 MB

  // stage 1: gather/pool + concat + pad (HBM-bound, ~1.1 GB traffic)
  pool_kernel<<<dim3(BATCH, NUM_TABLES), WIDTH, 0, stream>>>(tables, cat, X);
  tail_kernel<<<BATCH, 16, 0, stream>>>(numerical, X);
  padw1_kernel<<<XK, 256, 0, stream>>>(W1, W1p);

  // stage 2: MLP via fp32 WMMA
  gemm_bias_act<<<dim3(512 / BN, BATCH / BM), 256, 0, stream>>>(X,  W1p, b1, H1, 512, XK,  1);
  gemm_bias_act<<<dim3(256 / BN, BATCH / BM), 256, 0, stream>>>(H1, W2,  b2, H2, 256, 512, 1);
  gemm_bias_act<<<dim3(128 / BN, BATCH / BM), 256, 0, stream>>>(H2, W3,  b3, H3, 128, 256, 1);

  // stage 3: final dot
  final_layer<<<BATCH / 8, 256, 0, stream>>>(H3, W4, b4, out);
}